// Q_Act_13176959664395
// MI455X (gfx1250) — compile-verified
//
#include <hip/hip_runtime.h>
#include <stdint.h>

#define AS1 __attribute__((address_space(1)))
#define AS3 __attribute__((address_space(3)))

typedef int v4i __attribute__((ext_vector_type(4)));

#ifndef __has_builtin
#define __has_builtin(x) 0
#endif

#if __has_builtin(__builtin_amdgcn_global_load_async_to_lds_b128)
#define ASYNC_LDS 1
#endif

// ---- problem constants (from setup_inputs) ----
#define BB 32
#define LL 2000
#define DD 768
#define NR 10
#define GG 200            // LL / NR
#define NBG (BB * GG)     // 6400 rows per step
#define QMAX 7.0f         // n_lv/2 - 1
#define OQMAX 127.0f      // n_lva_o/2 - 1
#define MASK_WORDS 24     // 768 bits

// ---------------- CDNA5 async global->LDS staging ----------------
__device__ __forceinline__ void stage_issue(const float* gp, float* ls, int t) {
#ifdef ASYNC_LDS
  // 768 floats = 192 lanes x 16B ; row base is 3072B-aligned so 16B alignment holds
  if (t < 192) {
    __builtin_amdgcn_global_load_async_to_lds_b128(
        (AS1 v4i*)(AS1 void*)(gp + 4 * t),
        (AS3 v4i*)(AS3 void*)(ls + 4 * t), 0, 0);
  }
#else
#pragma unroll
  for (int c = 0; c < 3; ++c) ls[c * 256 + t] = gp[c * 256 + t];
#endif
}

__device__ __forceinline__ void stage_wait() {
#ifdef ASYNC_LDS
#if __has_builtin(__builtin_amdgcn_s_wait_asynccnt)
  __builtin_amdgcn_s_wait_asynccnt(0);
#else
  asm volatile("s_wait_asynccnt 0" ::: "memory");
#endif
#endif
  __syncthreads();
}

__device__ __forceinline__ uint32_t ballot32(bool p) {
#if __has_builtin(__builtin_amdgcn_ballot_w32)
  return __builtin_amdgcn_ballot_w32(p);
#else
  return (uint32_t)__ballot(p);
#endif
}

__device__ __forceinline__ float wave_sum(float v) {
#pragma unroll
  for (int off = 16; off > 0; off >>= 1) v += __shfl_xor(v, off, 32);
  return v;
}
__device__ __forceinline__ float wave_max(float v) {
#pragma unroll
  for (int off = 16; off > 0; off >>= 1) v = fmaxf(v, __shfl_xor(v, off, 32));
  return v;
}

// ---------------- kernels ----------------
// Phase 1 first in the file so the disasm snippet shows the async DMA prologue.
// masked sum / sumsq / max over D with OLD mask; set global flag.
__global__ __launch_bounds__(256) void k_stats(const float* __restrict__ x,
                                               const float* __restrict__ ss,
                                               const uint32_t* __restrict__ omask,
                                               float* __restrict__ stats,
                                               uint32_t* __restrict__ flags,
                                               int step) {
  const int t = threadIdx.x, bg = blockIdx.x;
  const int b = bg / GG, g = bg % GG;
  const int l = g * NR + step;
  const float* xp = x + ((size_t)b * LL + l) * DD;

  __shared__ __align__(16) float xs[DD];
  stage_issue(xp, xs, t);

  const int wave = t >> 5, lane = t & 31;
  uint32_t mw[3];
#pragma unroll
  for (int c = 0; c < 3; ++c) mw[c] = omask[bg * MASK_WORDS + c * 8 + wave];

  stage_wait();

  float s1 = 0.f, s2 = 0.f, mx = 0.f;
#pragma unroll
  for (int c = 0; c < 3; ++c) {
    bool o = (mw[c] >> lane) & 1u;
    float a = o ? 0.f : fabsf(xs[c * 256 + t]);
    s1 += a;
    s2 += a * a;
    mx = fmaxf(mx, a);
  }
  s1 = wave_sum(s1);
  s2 = wave_sum(s2);
  mx = wave_max(mx);

  __shared__ float r1[8], r2[8], rm[8];
  if (lane == 0) { r1[wave] = s1; r2[wave] = s2; rm[wave] = mx; }
  __syncthreads();
  if (t == 0) {
    float S1 = 0.f, S2 = 0.f, M = 0.f;
#pragma unroll
    for (int w = 0; w < 8; ++w) { S1 += r1[w]; S2 += r2[w]; M = fmaxf(M, rm[w]); }
    float4 st; st.x = S1; st.y = S2; st.z = M; st.w = 0.f;
    ((float4*)stats)[bg] = st;
    if (M > ss[l] * QMAX) atomicOr(&flags[step], 1u);  // dyn_s > static_s
  }
}

// Phase 2: conditional mask update, quantize inner + outlier parts, write y.
__global__ __launch_bounds__(256) void k_finish(const float* __restrict__ x,
                                                const float* __restrict__ ss,
                                                uint32_t* __restrict__ omask,
                                                const float* __restrict__ stats,
                                                const uint32_t* __restrict__ flags,
                                                float* __restrict__ y,
                                                int step) {
  const int t = threadIdx.x, bg = blockIdx.x;
  const int b = bg / GG, g = bg % GG;
  const int l = g * NR + step;
  const size_t base = ((size_t)b * LL + l) * (size_t)DD;
  const float* xp = x + base;

  __shared__ __align__(16) float xs[DD];
  stage_issue(xp, xs, t);

  const int wave = t >> 5, lane = t & 31;
  uint32_t mw[3];
#pragma unroll
  for (int c = 0; c < 3; ++c) mw[c] = omask[bg * MASK_WORDS + c * 8 + wave];
  const bool cond = (flags[step] != 0u);

  float thr = 0.f;
  if (cond) {
    float4 st = ((const float4*)stats)[bg];
    float mean = st.x * (1.0f / (float)DD);
    float var = (st.y - st.x * st.x * (1.0f / (float)DD)) * (1.0f / (float)(DD - 1));
    thr = mean + 3.0f * sqrtf(fmaxf(var, 0.f));
  }

  stage_wait();

  float xv[3];
  bool on[3];
  float mo = 0.f;
#pragma unroll
  for (int c = 0; c < 3; ++c) {
    bool o = (mw[c] >> lane) & 1u;
    float v = xs[c * 256 + t];
    // a = o ? 0 : |v| ; o_new = o || (a > thr)  ==  o || (|v| > thr)  (thr >= 0)
    bool nn = cond ? (o || (fabsf(v) > thr)) : o;
    xv[c] = v;
    on[c] = nn;
    if (nn) mo = fmaxf(mo, fabsf(v));
    if (cond) {
      uint32_t nwm = ballot32(nn);  // wave32 bitmask = exactly one mask dword
      if (lane == 0) omask[bg * MASK_WORDS + c * 8 + wave] = nwm;
    }
  }

  mo = wave_max(mo);
  __shared__ float rm2[8];
  __shared__ float os_sh;
  if (lane == 0) rm2[wave] = mo;
  __syncthreads();
  if (t == 0) {
    float M = 0.f;
#pragma unroll
    for (int w = 0; w < 8; ++w) M = fmaxf(M, rm2[w]);
    os_sh = fmaxf(M, 1e-8f) * (1.0f / OQMAX);
  }
  __syncthreads();

  const float o_s = os_sh;
  const float s = ss[l];
  float* yp = y + base;
#pragma unroll
  for (int c = 0; c < 3; ++c) {
    float v = xv[c];
    float inl = on[c] ? 0.f : v;
    float outl = on[c] ? v : 0.f;
    float q = fminf(fmaxf(rintf(inl / s), -QMAX), QMAX);     // RNE == jnp.round
    float oq = fminf(fmaxf(rintf(outl / o_s), -OQMAX), OQMAX);
    __builtin_nontemporal_store(q * s + oq * o_s, yp + c * 256 + t);  // keep x resident in L2
  }
}

__global__ __launch_bounds__(256) void k_init(uint32_t* flags, uint32_t* omask) {
  int i = blockIdx.x * blockDim.x + threadIdx.x;
  if (i < 16) flags[i] = 0u;
  const int n = NBG * MASK_WORDS;
  for (int j = i; j < n; j += gridDim.x * blockDim.x) omask[j] = 0u;
}

// ---------------- launch ----------------
extern "C" void kernel_launch(void* const* d_in, const int* in_sizes, int n_in,
                              void* d_out, int out_size, void* d_ws, size_t ws_size,
                              hipStream_t stream) {
  const float* x = (const float*)d_in[0];
  const float* ss = (const float*)d_in[1];
  float* y = (float*)d_out;

  // workspace layout: [flags: 16 u32][stats: NBG * float4][omask: NBG * 24 u32]
  uint32_t* flags = (uint32_t*)d_ws;
  float* stats = (float*)((char*)d_ws + 64);
  uint32_t* omask = (uint32_t*)((char*)d_ws + 64 + (size_t)NBG * 16);

  k_init<<<600, 256, 0, stream>>>(flags, omask);
  for (int i = 0; i < NR; ++i) {
    k_stats<<<NBG, 256, 0, stream>>>(x, ss, omask, stats, flags, i);
    k_finish<<<NBG, 256, 0, stream>>>(x, ss, omask, stats, flags, y, i);
  }
}